// time_gat_4080218931413
// MI455X (gfx1250) — compile-verified
//
#include <hip/hip_runtime.h>
#include <math.h>

typedef __attribute__((ext_vector_type(2))) float v2f;
typedef __attribute__((ext_vector_type(8))) float v8f;

#define N_NODES 8192
#define HDIM    128
#define ALPHA_L 0.2f
// Rows with t_i = 2^-(8191-i) representable (even subnormal) in f32: 8191-i <= 149.
// Take 160 active rows for margin; the extras have t_i == 0.0f and contribute 0.
#define NACT 160
#define I0   (N_NODES - NACT)
#define TSUM 2.0f   // sum_{k=0}^{151} 2^-k rounds to exactly 2.0f

// ---------------------------------------------------------------------------
// Kernel 1: h = inp @ W via v_wmma_f32_16x16x4_f32. One wave = one 16x16 tile.
// 512 m-tiles x 8 n-tiles = 4096 waves.
// ---------------------------------------------------------------------------
__global__ __launch_bounds__(256) void k_gemm_h(const float* __restrict__ inp,
                                                const float* __restrict__ Wm,
                                                float* __restrict__ h) {
  const int wave = (blockIdx.x * blockDim.x + threadIdx.x) >> 5;
  const int lane = threadIdx.x & 31;
  const int tile_m = wave >> 3;    // 0..511
  const int tile_n = wave & 7;     // 0..7
  const int m0 = tile_m * 16, n0 = tile_n * 16;
  const int half = lane >> 4;      // K-half select
  const int lm = lane & 15;        // M (A) / N (B,C)

  v8f acc = {};
  // A frag base: inp[m0+lm][2*half + k], consecutive pair per lane
  const float* arow = inp + (m0 + lm) * HDIM + 2 * half;
  // B frag base: W[2*half + k][n0+lm]
  const float* bcol = Wm + (2 * half) * HDIM + n0 + lm;

  for (int k = 0; k < HDIM; k += 4) {
    v2f a; a.x = arow[k];        a.y = arow[k + 1];
    v2f b; b.x = bcol[k * HDIM]; b.y = bcol[(k + 1) * HDIM];
    acc = __builtin_amdgcn_wmma_f32_16x16x4_f32(
        /*neg_a=*/false, a, /*neg_b=*/false, b,
        /*c_mod=*/(short)0, acc, /*reuse_a=*/false, /*reuse_b=*/false);
  }
  // D layout: VGPR r, lane -> (M = r + 8*half, N = lm)
  float* out = h + (m0 + 8 * half) * HDIM + n0 + lm;
#pragma unroll
  for (int r = 0; r < 8; ++r) out[r * HDIM] = acc[r];
}

// ---------------------------------------------------------------------------
// Kernel 2: f1[i] = h[i,:]@a[0:128], f2[i] = h[i,:]@a[128:256]. One wave/row.
// ---------------------------------------------------------------------------
__global__ __launch_bounds__(256) void k_f1f2(const float* __restrict__ h,
                                              const float* __restrict__ av,
                                              float* __restrict__ f1,
                                              float* __restrict__ f2) {
  const int row  = (blockIdx.x * blockDim.x + threadIdx.x) >> 5;
  const int lane = threadIdx.x & 31;
  if (row >= N_NODES) return;
  const float* hr = h + row * HDIM;
  float s1 = 0.f, s2 = 0.f;
#pragma unroll
  for (int c = 0; c < HDIM; c += 32) {
    const float hv = hr[c + lane];
    s1 += hv * av[c + lane];
    s2 += hv * av[HDIM + c + lane];
  }
#pragma unroll
  for (int off = 16; off > 0; off >>= 1) {
    s1 += __shfl_xor(s1, off, 32);
    s2 += __shfl_xor(s2, off, 32);
  }
  if (lane == 0) { f1[row] = s1; f2[row] = s2; }
}

// ---------------------------------------------------------------------------
// Kernel 3: per active row i: m_i = max masked e_ij, l_i = sum exp(e - m_i).
// One 256-thread block per active row; coalesced sweep of A-row.
// ---------------------------------------------------------------------------
__global__ __launch_bounds__(256) void k_rowstats(const int* __restrict__ A,
                                                  const float* __restrict__ f1,
                                                  const float* __restrict__ f2,
                                                  float* __restrict__ mrow,
                                                  float* __restrict__ lrow) {
  const int row = I0 + blockIdx.x;
  const int tid = threadIdx.x;
  __shared__ float red[256];
  const int* Ar = A + (long long)row * N_NODES;
  const float f1i = f1[row];

  float m = -3.0e38f;
  for (int j = tid; j < N_NODES; j += 256) {
    if (Ar[j] != 0 || j == row) {
      float e = f1i + f2[j];
      e = (e >= 0.f) ? e : ALPHA_L * e;
      m = fmaxf(m, e);
    }
  }
  red[tid] = m; __syncthreads();
  for (int s = 128; s > 0; s >>= 1) {
    if (tid < s) red[tid] = fmaxf(red[tid], red[tid + s]);
    __syncthreads();
  }
  const float mi = red[0];
  __syncthreads();

  float l = 0.f;
  for (int j = tid; j < N_NODES; j += 256) {
    if (Ar[j] != 0 || j == row) {
      float e = f1i + f2[j];
      e = (e >= 0.f) ? e : ALPHA_L * e;
      l += expf(e - mi);
    }
  }
  red[tid] = l; __syncthreads();
  for (int s = 128; s > 0; s >>= 1) {
    if (tid < s) red[tid] += red[tid + s];
    __syncthreads();
  }
  if (tid == 0) { mrow[blockIdx.x] = mi; lrow[blockIdx.x] = red[0]; }
}

// ---------------------------------------------------------------------------
// Kernel 4: w_j = sum_{i active} (t_i/(TSUM*l_i)) * exp(e_ij - m_i) * mask_ij.
// Thread per column j; loop over active rows is coalesced across j.
// ---------------------------------------------------------------------------
__global__ __launch_bounds__(256) void k_wj(const int* __restrict__ A,
                                            const float* __restrict__ f1,
                                            const float* __restrict__ f2,
                                            const float* __restrict__ mrow,
                                            const float* __restrict__ lrow,
                                            float* __restrict__ wcol) {
  __shared__ float sc[NACT], sm[NACT], sf1[NACT];
  const int tid = threadIdx.x;
  for (int il = tid; il < NACT; il += 256) {
    const int row = I0 + il;
    const float t = exp2f(-(float)(N_NODES - 1 - row));  // 0.0f for inactive tail
    sc[il]  = t / (TSUM * lrow[il]);
    sm[il]  = mrow[il];
    sf1[il] = f1[row];
  }
  __syncthreads();

  const int j = blockIdx.x * 256 + tid;
  const float f2j = f2[j];
  float w = 0.f;
  for (int il = 0; il < NACT; ++il) {
    const int row = I0 + il;
    const int aij = A[(long long)row * N_NODES + j];
    if (aij != 0 || j == row) {
      float e = sf1[il] + f2j;
      e = (e >= 0.f) ? e : ALPHA_L * e;
      w += sc[il] * expf(e - sm[il]);
    }
  }
  wcol[j] = w;
}

// ---------------------------------------------------------------------------
// Kernel 5a: zero the 128-float accumulator.
// ---------------------------------------------------------------------------
__global__ void k_zero(float* __restrict__ acc) { acc[threadIdx.x] = 0.f; }

// ---------------------------------------------------------------------------
// Kernel 5b: h_t[e] = sum_j w_j * h[j][e]. Block handles a 128-row j slab,
// thread e-major (coalesced h reads), atomic combine into acc.
// ---------------------------------------------------------------------------
__global__ __launch_bounds__(256) void k_pool(const float* __restrict__ h,
                                              const float* __restrict__ wcol,
                                              float* __restrict__ acc) {
  __shared__ float red[256];
  const int tid = threadIdx.x;
  const int e = tid & (HDIM - 1);
  const int g = tid >> 7;            // 0..1
  const int jbase = blockIdx.x * 128;
  float s = 0.f;
  for (int jj = g; jj < 128; jj += 2) {
    const int j = jbase + jj;
    s += wcol[j] * h[j * HDIM + e];
  }
  red[tid] = s; __syncthreads();
  if (g == 0) atomicAdd(&acc[e], red[tid] + red[tid + 128]);
}

// ---------------------------------------------------------------------------
// Kernel 6: out = ELU(h_t).
// ---------------------------------------------------------------------------
__global__ void k_elu(const float* __restrict__ acc, float* __restrict__ out) {
  const int e = threadIdx.x;
  const float x = acc[e];
  out[e] = (x > 0.f) ? x : expm1f(x);
}

// ---------------------------------------------------------------------------
extern "C" void kernel_launch(void* const* d_in, const int* in_sizes, int n_in,
                              void* d_out, int out_size, void* d_ws, size_t ws_size,
                              hipStream_t stream) {
  const float* inp = (const float*)d_in[0];   // [8192,128]
  const int*   A   = (const int*)  d_in[1];   // [8192,8192]
  const float* Wm  = (const float*)d_in[2];   // [128,128]
  const float* av  = (const float*)d_in[3];   // [256]
  float* out = (float*)d_out;                 // [128]

  float* ws = (float*)d_ws;
  float* h    = ws;                                  // 8192*128
  float* f1   = h    + N_NODES * HDIM;               // 8192
  float* f2   = f1   + N_NODES;                      // 8192
  float* mrow = f2   + N_NODES;                      // NACT
  float* lrow = mrow + NACT;                         // NACT
  float* wcol = lrow + NACT;                         // 8192
  float* acc  = wcol + N_NODES;                      // 128

  // 1) h = inp @ W  (WMMA f32): 4096 waves = 512 blocks * 8 waves
  k_gemm_h<<<512, 256, 0, stream>>>(inp, Wm, h);
  // 2) f1/f2: 8192 waves = 1024 blocks * 8 waves
  k_f1f2<<<1024, 256, 0, stream>>>(h, av, f1, f2);
  // 3) softmax stats for active rows
  k_rowstats<<<NACT, 256, 0, stream>>>(A, f1, f2, mrow, lrow);
  // 4) pooled column weights
  k_wj<<<N_NODES / 256, 256, 0, stream>>>(A, f1, f2, mrow, lrow, wcol);
  // 5) h_t = w @ h
  k_zero<<<1, HDIM, 0, stream>>>(acc);
  k_pool<<<N_NODES / 128, 256, 0, stream>>>(h, wcol, acc);
  // 6) ELU
  k_elu<<<1, HDIM, 0, stream>>>(acc, out);
}